// SpatialTransformer_32770600468590
// MI455X (gfx1250) — compile-verified
//
#include <hip/hip_runtime.h>

// 3-D spatial transformer (trilinear warp), B=2, D=160, C=1.
// Memory-bound gather: ~164 MB total traffic -> ~7 us floor at 23.3 TB/s.
// WMMA does not apply (0.25 GFLOP total); optimize the data-movement path:
// coalesced b96 NT loads of trf, float2 (b64) gathers of img (z-pairs are
// contiguous), NT store of out, and a prefetch of the identity-grid img line.

namespace {
constexpr int Dm  = 160;
constexpr int DD  = Dm * Dm;         // 25,600
constexpr int DDD = Dm * DD;         // 4,096,000

typedef float vf2 __attribute__((ext_vector_type(2)));
// 4-byte-aligned view so the compiler may merge the z-pair into one b64 load
// (global memory on gfx1250 only needs DWORD alignment in DWORD mode).
typedef vf2 __attribute__((aligned(4))) uvf2;

__device__ __forceinline__ vf2 load2(const float* p) {
  return *reinterpret_cast<const uvf2*>(p);
}
} // namespace

__global__ __launch_bounds__(Dm)
void st3d_warp_kernel(const float* __restrict__ img,
                      const float* __restrict__ trf,
                      float* __restrict__ out) {
  const int z = threadIdx.x;      // unit-stride spatial dim (stride 1)
  const int y = blockIdx.x;       // stride D
  const int x = blockIdx.y;       // stride D*D
  const int b = blockIdx.z;

  const int vox = ((b * Dm + x) * Dm + y) * Dm + z;
  const float* __restrict__ ib = img + b * DDD;

  // Warm the cache line(s) around the identity-grid location: the gather
  // targets are within a few voxels of here (trf ~ N(0,1)). Overlaps the
  // line fill with the trf load + address math. -> global_prefetch_b8
  __builtin_prefetch(ib + x * DD + y * Dm + z, 0, 0);

  // Stream-once displacement triplet: non-temporal so L2 stays img-resident.
  const float* t = trf + vox * 3;
  const float tx = __builtin_nontemporal_load(t + 0);
  const float ty = __builtin_nontemporal_load(t + 1);
  const float tz = __builtin_nontemporal_load(t + 2);

  // Clamp warped coordinates to [0, D-1] (reference: fill_value=None).
  const float fmax = (float)(Dm - 1);
  const float fx = fminf(fmaxf((float)x + tx, 0.0f), fmax);
  const float fy = fminf(fmaxf((float)y + ty, 0.0f), fmax);
  const float fz = fminf(fmaxf((float)z + tz, 0.0f), fmax);

  const float x0f = floorf(fx), y0f = floorf(fy), z0f = floorf(fz);
  const int x0 = (int)x0f, y0 = (int)y0f, z0 = (int)z0f;
  const float wx1 = fx - x0f, wy1 = fy - y0f, wz1 = fz - z0f;
  const float wx0 = 1.0f - wx1, wy0 = 1.0f - wy1, wz0 = 1.0f - wz1;

  const int x1 = min(x0 + 1, Dm - 1);
  const int y1 = min(y0 + 1, Dm - 1);
  // z-corners are contiguous: load the pair at z0L = min(z0, D-2).
  // If z0 == D-1 the clamp guarantees wz1 == 0, and the shifted pair's .y
  // element *is* img[..., D-1], so the edge case stays exact.
  const int  z0L = min(z0, Dm - 2);
  const bool zlo = (z0 == z0L);

  const int rx0 = x0 * DD, rx1 = x1 * DD;
  const int ry0 = y0 * Dm, ry1 = y1 * Dm;

  const vf2 p00 = load2(ib + rx0 + ry0 + z0L);
  const vf2 p01 = load2(ib + rx0 + ry1 + z0L);
  const vf2 p10 = load2(ib + rx1 + ry0 + z0L);
  const vf2 p11 = load2(ib + rx1 + ry1 + z0L);

  // Lerp along z (branchless edge handling), then y, then x.
  const float c00 = zlo ? fmaf(wz0, p00.x, wz1 * p00.y) : p00.y;
  const float c01 = zlo ? fmaf(wz0, p01.x, wz1 * p01.y) : p01.y;
  const float c10 = zlo ? fmaf(wz0, p10.x, wz1 * p10.y) : p10.y;
  const float c11 = zlo ? fmaf(wz0, p11.x, wz1 * p11.y) : p11.y;

  const float c0 = fmaf(wy0, c00, wy1 * c01);
  const float c1 = fmaf(wy0, c10, wy1 * c11);
  const float r  = fmaf(wx0, c0,  wx1 * c1);

  __builtin_nontemporal_store(r, out + vox);   // stream-once output, th:NT
}

extern "C" void kernel_launch(void* const* d_in, const int* in_sizes, int n_in,
                              void* d_out, int out_size, void* d_ws, size_t ws_size,
                              hipStream_t stream) {
  const float* img = (const float*)d_in[0];  // [B, D, D, D, 1] f32
  const float* trf = (const float*)d_in[1];  // [B, D, D, D, 3] f32
  float* out = (float*)d_out;                // [B, D, D, D, 1] f32

  const int B = in_sizes[0] / DDD;           // = 2 for the reference shapes
  dim3 grid(Dm, Dm, B);                      // (y, x, b)
  dim3 block(Dm);                            // z row: 5 wave32s
  st3d_warp_kernel<<<grid, block, 0, stream>>>(img, trf, out);
}